// ChebConv_49383533969583
// MI455X (gfx1250) — compile-verified
//
#include <hip/hip_runtime.h>
#include <hip/hip_bf16.h>

typedef __attribute__((ext_vector_type(2))) float v2f;
typedef __attribute__((ext_vector_type(8))) float v8f;

#define FEAT 64
#define KDIM 192  // 3 * FEAT
#define KPAIRS (KDIM / 2)

// ---------------------------------------------------------------------------
// 1) in-degree: deg[dst[e]] += 1  (hardware f32 atomics into L2)
// ---------------------------------------------------------------------------
__global__ __launch_bounds__(256) void cheb_deg(const int* __restrict__ dst,
                                                float* __restrict__ deg, int nedges) {
  int e = blockIdx.x * 256 + threadIdx.x;
  if (e < nedges) unsafeAtomicAdd(&deg[dst[e]], 1.0f);
}

// ---------------------------------------------------------------------------
// 2) dinv[i] = rsqrt(max(deg,1))
// ---------------------------------------------------------------------------
__global__ __launch_bounds__(256) void cheb_dinv(const float* __restrict__ deg,
                                                 float* __restrict__ dinv, int n) {
  int i = blockIdx.x * 256 + threadIdx.x;
  if (i < n) {
    float d = deg[i];
    d = fmaxf(d, 1.0f);
    dinv[i] = rsqrtf(d);
  }
}

// ---------------------------------------------------------------------------
// 3) SpMM scatter: acc[dst] += X[src] * dinv[src]
//    one wave32 per edge, float2 per lane (64 feats), f32 atomics (L2 resident)
// ---------------------------------------------------------------------------
__global__ __launch_bounds__(256) void cheb_spmm(const float* __restrict__ X,
                                                 const float* __restrict__ dinv,
                                                 const int* __restrict__ src,
                                                 const int* __restrict__ dst,
                                                 float* __restrict__ acc, int nedges) {
  int e = blockIdx.x * 8 + (threadIdx.x >> 5);
  if (e >= nedges) return;
  int lane = threadIdx.x & 31;
  int s = src[e];
  int d = dst[e];
  float sc = dinv[s];
  const float2* xr = (const float2*)(X + (size_t)s * FEAT);
  float2 v = xr[lane];
  float* o = acc + (size_t)d * FEAT + lane * 2;
  unsafeAtomicAdd(o,     v.x * sc);
  unsafeAtomicAdd(o + 1, v.y * sc);
}

// ---------------------------------------------------------------------------
// 4a) X1 = -acc * dinv[row]          (re_norm == 1)
// ---------------------------------------------------------------------------
__global__ __launch_bounds__(256) void cheb_fin1(float* __restrict__ buf,
                                                 const float* __restrict__ dinv, int n) {
  int i = blockIdx.x * 256 + threadIdx.x;
  if (i < n) buf[i] = -buf[i] * dinv[i >> 6];
}

// 4b) X2 = -2*acc*dinv[row] - X0
__global__ __launch_bounds__(256) void cheb_fin2(float* __restrict__ buf,
                                                 const float* __restrict__ dinv,
                                                 const float* __restrict__ X0, int n) {
  int i = blockIdx.x * 256 + threadIdx.x;
  if (i < n) buf[i] = -2.0f * buf[i] * dinv[i >> 6] - X0[i];
}

// ---------------------------------------------------------------------------
// 5) out = relu([X0|X1|X2] @ W + b) via V_WMMA_F32_16X16X4_F32
//    one wave per 16-row tile; 4 accumulators = all 64 output cols.
//    W is staged in LDS pre-packed as K-pairs so each B fragment is ONE
//    aligned ds_load_b64 landing directly in the WMMA operand pair
//    (eliminates the v_dual_mov shuffles seen in the previous build).
// ---------------------------------------------------------------------------
__global__ __launch_bounds__(256) void cheb_gemm(const float* __restrict__ X0,
                                                 const float* __restrict__ X1,
                                                 const float* __restrict__ X2,
                                                 const float* __restrict__ W,
                                                 const float* __restrict__ bias,
                                                 float* __restrict__ out, int nrows) {
  // Ws[kpair * FEAT + col] = { W[2*kpair][col], W[2*kpair+1][col] }   (48 KB)
  __shared__ v2f Ws[KPAIRS * FEAT];
  for (int i = threadIdx.x; i < KPAIRS * FEAT; i += 256) {
    const int kp = i >> 6;   // pair index 0..95
    const int c  = i & 63;
    v2f p;
    p.x = W[(2 * kp) * FEAT + c];
    p.y = W[(2 * kp + 1) * FEAT + c];
    Ws[i] = p;
  }
  __syncthreads();

  const int wave = threadIdx.x >> 5;
  const int lane = threadIdx.x & 31;
  const int tile = blockIdx.x * 8 + wave;
  const int row0 = tile * 16;
  if (row0 >= nrows) return;  // whole wave exits together (EXEC stays all-1 for WMMA)

  // A-fragment layout (16x4 f32): lanes 0-15 -> K={0,1}, lanes 16-31 -> K={2,3}
  const int arow = lane & 15;
  const int kb   = (lane >> 4) * 2;  // 0 or 2
  const int col  = lane & 15;

  v8f acc[4] = {v8f{}, v8f{}, v8f{}, v8f{}};
  const float* Xs[3] = {X0, X1, X2};

#pragma unroll
  for (int seg = 0; seg < 3; ++seg) {
    const float* xrow = Xs[seg] + (size_t)(row0 + arow) * FEAT;
#pragma unroll 4
    for (int k = 0; k < FEAT; k += 4) {
      // one aligned global_load_b64 for the A fragment
      const v2f a = *(const v2f*)(xrow + k + kb);
      const int kw = seg * FEAT + k + kb;           // even
      const v2f* brow = Ws + (kw >> 1) * FEAT + col;
#pragma unroll
      for (int t = 0; t < 4; ++t) {
        const v2f bf = brow[t * 16];                // one ds_load_b64
        acc[t] = __builtin_amdgcn_wmma_f32_16x16x4_f32(
            false, a, false, bf, (short)0, acc[t], false, false);
      }
    }
  }

  // C/D layout: VGPR v -> row v (lanes 0-15) / row v+8 (lanes 16-31), col = lane&15
  const int mbase = (lane >> 4) * 8;
#pragma unroll
  for (int t = 0; t < 4; ++t) {
    const int c = t * 16 + col;
    const float bv = bias[c];
#pragma unroll
    for (int v = 0; v < 8; ++v) {
      float val = acc[t][v] + bv;
      out[(size_t)(row0 + mbase + v) * FEAT + c] = val > 0.0f ? val : 0.0f;
    }
  }
}

// ---------------------------------------------------------------------------
// launch: feat, src, dst, W, b  ->  out [N,64] f32
// ---------------------------------------------------------------------------
extern "C" void kernel_launch(void* const* d_in, const int* in_sizes, int n_in,
                              void* d_out, int out_size, void* d_ws, size_t ws_size,
                              hipStream_t stream) {
  const float* feat = (const float*)d_in[0];
  const int*   src  = (const int*)d_in[1];
  const int*   dst  = (const int*)d_in[2];
  const float* W    = (const float*)d_in[3];
  const float* bias = (const float*)d_in[4];
  float* out = (float*)d_out;

  const int N = in_sizes[0] / FEAT;
  const int E = in_sizes[1];

  float* ws   = (float*)d_ws;
  float* deg  = ws;                         // N
  float* dinv = ws + N;                     // N
  float* buf1 = ws + 2 * (size_t)N;         // N*64  (X1)
  float* buf2 = buf1 + (size_t)N * FEAT;    // N*64  (X2)

  const int NF = N * FEAT;

  // degrees -> dinv
  hipMemsetAsync(deg, 0, (size_t)N * sizeof(float), stream);
  cheb_deg<<<(E + 255) / 256, 256, 0, stream>>>(dst, deg, E);
  cheb_dinv<<<(N + 255) / 256, 256, 0, stream>>>(deg, dinv, N);

  // X1 = -L(X0)
  hipMemsetAsync(buf1, 0, (size_t)NF * sizeof(float), stream);
  cheb_spmm<<<(E + 7) / 8, 256, 0, stream>>>(feat, dinv, src, dst, buf1, E);
  cheb_fin1<<<(NF + 255) / 256, 256, 0, stream>>>(buf1, dinv, NF);

  // X2 = -2*L(X1) - X0
  hipMemsetAsync(buf2, 0, (size_t)NF * sizeof(float), stream);
  cheb_spmm<<<(E + 7) / 8, 256, 0, stream>>>(buf1, dinv, src, dst, buf2, E);
  cheb_fin2<<<(NF + 255) / 256, 256, 0, stream>>>(buf2, dinv, feat, NF);

  // out = relu([X0|X1|X2] @ W + b)  -- fp32 WMMA
  const int ntiles = (N + 15) / 16;
  cheb_gemm<<<(ntiles + 7) / 8, 256, 0, stream>>>(feat, buf1, buf2, W, bias, out, N);
}